// BayesianLinearModel_91233695302379
// MI455X (gfx1250) — compile-verified
//
#include <hip/hip_runtime.h>
#include <hip/hip_bf16.h>
#include <math.h>

// Problem constants (match reference)
#define B_DIM 2048
#define K_DIM 8
#define O_DIM 256
#define I_DIM 512
#define TINYV 1e-10f

typedef __attribute__((ext_vector_type(2))) float v2f;
typedef __attribute__((ext_vector_type(4))) float v4f;
typedef __attribute__((ext_vector_type(8))) float v8f;

// ---------------------------------------------------------------------------
// Kernel 1: per-row gumbel-max component selection.
// mix is numerically one-hot in the reference forward pass; argmax of
// (mix_weights[k] + gumbel(u))/tau == argmax of mix_weights[k] + gumbel(u).
// ---------------------------------------------------------------------------
__global__ void select_k_kernel(const float* __restrict__ mw,
                                const float* __restrict__ u,
                                int* __restrict__ kidx) {
    int b = blockIdx.x * blockDim.x + threadIdx.x;
    if (b >= B_DIM) return;
    float best = -INFINITY;
    int bi = 0;
#pragma unroll
    for (int k = 0; k < K_DIM; ++k) {
        float uu = u[b * K_DIM + k];
        float g  = -logf(-logf(uu + TINYV) + TINYV);
        float s  = mw[k] + g;  // tau==1, and division is monotone anyway
        if (s > best) { best = s; bi = k; }  // strict > keeps first max (argmax semantics)
    }
    kidx[b] = bi;
}

// ---------------------------------------------------------------------------
// Kernel 2: sigma = exp(log_sigma), done once (K*O*I = 1M elements, 4 MB).
// ---------------------------------------------------------------------------
__global__ void exp_sigma_kernel(const float* __restrict__ ls,
                                 float* __restrict__ sigma, int n) {
    int i = blockIdx.x * blockDim.x + threadIdx.x;
    if (i < n) sigma[i] = __expf(ls[i]);
}

// ---------------------------------------------------------------------------
// Kernel 3: Y[k,b,o] = sum_i X[b,i] * mean[k,o,i]  via V_WMMA_F32_16X16X4_F32.
// One wave32 per 16(b) x 16(o) tile, loop i in steps of 4.
// A (16x4 f32): lanes 0-15 hold M=lane, K=0..1 in v[0..1]; lanes 16-31 hold
// M=lane-16, K=2..3.  B (4x16) mirrors with N=lane%16.  C/D: 8 VGPRs,
// lane<16 -> M=r, lane>=16 -> M=r+8, N=lane%16.
// ---------------------------------------------------------------------------
__global__ void gemm_y_kernel(const float* __restrict__ X,
                              const float* __restrict__ mean,
                              float* __restrict__ Y) {
    const int b0   = blockIdx.x * 16;
    const int o0   = blockIdx.y * 16;
    const int k    = blockIdx.z;
    const int lane = threadIdx.x;       // 0..31 (one wave per block)
    const int half = lane >> 4;          // 0: K=0..1, 1: K=2..3
    const int l16  = lane & 15;

    const float2* Arow = (const float2*)(X + (size_t)(b0 + l16) * I_DIM);
    const float2* Brow = (const float2*)(mean + ((size_t)k * O_DIM + (o0 + l16)) * I_DIM);

    v8f c = {};
#pragma unroll 4
    for (int i0 = 0; i0 < I_DIM; i0 += 4) {
        const int p = (i0 >> 1) + half;  // float2 index: i0/2 (+1 for upper half-lanes)
        float2 af = Arow[p];
        float2 bf = Brow[p];
        v2f a; a.x = af.x; a.y = af.y;
        v2f b; b.x = bf.x; b.y = bf.y;
        c = __builtin_amdgcn_wmma_f32_16x16x4_f32(
            /*neg_a=*/false, a, /*neg_b=*/false, b,
            /*c_mod=*/(short)0, c, /*reuse_a=*/false, /*reuse_b=*/false);
    }

    float* Yt = Y + ((size_t)k * B_DIM + b0) * O_DIM + o0;
#pragma unroll
    for (int r = 0; r < 8; ++r) {
        int row = r + half * 8;
        Yt[(size_t)row * O_DIM + l16] = c[r];
    }
}

// ---------------------------------------------------------------------------
// Kernel 4 (dominant, bandwidth-bound): one block per batch row b.
// out[b,o] = Y[k*,b,o] + sum_i X[b,i] * sigma[k*,o,i] * eps[b,o,i]
// eps (1.07 GB) is streamed with non-temporal loads so sigma/mean/X/Y stay
// L2-resident.  8 waves per block; each wave owns 32 'o' rows; per row the
// wave does 4 fully-coalesced 512B float4 load iterations, then a wave32
// shuffle reduction.
// ---------------------------------------------------------------------------
__global__ __launch_bounds__(256)
void stream_kernel(const float* __restrict__ X,
                   const float* __restrict__ eps,
                   const float* __restrict__ sigma,
                   const float* __restrict__ Y,
                   const int* __restrict__ kidx,
                   float* __restrict__ out) {
    const int b    = blockIdx.x;
    const int tid  = threadIdx.x;
    const int wave = tid >> 5;
    const int lane = tid & 31;

    __shared__ float sX[I_DIM];
    for (int i = tid; i < I_DIM; i += 256) sX[i] = X[(size_t)b * I_DIM + i];
    __syncthreads();

    const int kb = kidx[b];
    const float* sig  = sigma + (size_t)kb * O_DIM * I_DIM;
    const float* epsb = eps + (size_t)b * O_DIM * I_DIM;
    const float* Ysel = Y + ((size_t)kb * B_DIM + b) * O_DIM;
    const v4f* x4 = (const v4f*)sX;

    for (int o = wave; o < O_DIM; o += 8) {
        const v4f* e4 = (const v4f*)(epsb + (size_t)o * I_DIM);
        const v4f* s4 = (const v4f*)(sig  + (size_t)o * I_DIM);
        float acc = 0.0f;
#pragma unroll
        for (int it = 0; it < I_DIM / 4; it += 32) {
            const int p = it + lane;
            v4f e = __builtin_nontemporal_load(&e4[p]);  // streaming: don't pollute L2
            v4f s = s4[p];
            v4f x = x4[p];
            acc = fmaf(x.x * s.x, e.x, acc);
            acc = fmaf(x.y * s.y, e.y, acc);
            acc = fmaf(x.z * s.z, e.z, acc);
            acc = fmaf(x.w * s.w, e.w, acc);
        }
        // wave32 reduction
#pragma unroll
        for (int off = 16; off > 0; off >>= 1)
            acc += __shfl_down(acc, off, 32);
        if (lane == 0)
            out[(size_t)b * O_DIM + o] = Ysel[o] + acc;
    }
}

// ---------------------------------------------------------------------------
// Launcher.  Workspace layout:
//   [0, 8K)        : int   kidx[B]
//   [8K, 8K+4M)    : float sigma[K*O*I]
//   [8K+4M, +16M)  : float Y[K*B*O]
// Total ~21 MB.
// ---------------------------------------------------------------------------
extern "C" void kernel_launch(void* const* d_in, const int* in_sizes, int n_in,
                              void* d_out, int out_size, void* d_ws, size_t ws_size,
                              hipStream_t stream) {
    (void)in_sizes; (void)n_in; (void)out_size; (void)ws_size;
    const float* X         = (const float*)d_in[0];
    const float* mix_w     = (const float*)d_in[1];
    const float* mean      = (const float*)d_in[2];
    const float* log_sigma = (const float*)d_in[3];
    const float* u_gumbel  = (const float*)d_in[4];
    const float* eps       = (const float*)d_in[5];
    float* out = (float*)d_out;

    char* wsb    = (char*)d_ws;
    int*   kidx  = (int*)wsb;
    float* sigma = (float*)(wsb + 8192);
    float* Y     = (float*)(wsb + 8192 + (size_t)K_DIM * O_DIM * I_DIM * sizeof(float));

    // 1) component selection (argmax of gumbel-perturbed logits)
    select_k_kernel<<<(B_DIM + 255) / 256, 256, 0, stream>>>(mix_w, u_gumbel, kidx);

    // 2) sigma = exp(log_sigma)
    const int nsig = K_DIM * O_DIM * I_DIM;
    exp_sigma_kernel<<<(nsig + 255) / 256, 256, 0, stream>>>(log_sigma, sigma, nsig);

    // 3) WMMA GEMMs: Y[k] = X * mean[k]^T for all k
    {
        dim3 grid(B_DIM / 16, O_DIM / 16, K_DIM);
        gemm_y_kernel<<<grid, 32, 0, stream>>>(X, mean, Y);
    }

    // 4) bandwidth-bound eps stream + final output
    stream_kernel<<<B_DIM, 256, 0, stream>>>(X, eps, sigma, Y, kidx, out);
}